// LinearTransformer_30588757082644
// MI455X (gfx1250) — compile-verified
//
#include <hip/hip_runtime.h>

// ---------------- problem constants ----------------
#define C_    512          // channels
#define N_    1024         // tokens per batch (H*W)
#define B_    16           // batch
#define H_    32           // spatial H (softmax rows per batch)
#define PB_   (C_ * N_)    // 524288 elements per batch (2^19)
#define FFW_  2048
#define EPS_F 1e-5f

// ---------------- WMMA types ----------------
typedef float v2f __attribute__((ext_vector_type(2)));
typedef float v8f __attribute__((ext_vector_type(8)));

static __device__ __forceinline__ v8f wmma4(v2f a, v2f b, v8f c) {
  // V_WMMA_F32_16X16X4_F32 : D = A(16x4) x B(4x16) + C(16x16), all fp32
  return __builtin_amdgcn_wmma_f32_16x16x4_f32(
      /*neg_a=*/false, a, /*neg_b=*/false, b,
      /*c_mod=*/(short)0, c, /*reuse_a=*/false, /*reuse_b=*/false);
}

// ============================================================
// Batched GEMM: D[b](O,N) = W(O,K) @ X[b](K,N) (+bias[o]) (+res[b])
// block = 128 threads (4 waves); each wave computes a 32x64 tile
// (2 M-tiles x 4 N-tiles of 16x16).  Workgroup covers 64 x 128.
// Grid: (O/64, N/128, B). O%64==0, N%128==0, K%8==0 required.
// ============================================================
__global__ __launch_bounds__(128)
void gemm_wmma_f32(const float* __restrict__ Wt, const float* __restrict__ X,
                   const float* __restrict__ bias, const float* __restrict__ res,
                   float* __restrict__ D, int O, int K, int N,
                   size_t xStride, size_t dStride)
{
  const int wave = threadIdx.x >> 5;
  const int lane = threadIdx.x & 31;
  const int lid  = lane & 15;   // column/row-within-16
  const int hi   = lane >> 4;   // lane group (selects K pair / M+8)

  const int m0 = blockIdx.x * 64 + (wave & 1) * 32;
  const int n0 = blockIdx.y * 128 + (wave >> 1) * 64;

  const float* Xb = X + (size_t)blockIdx.z * xStride;
  float*       Db = D + (size_t)blockIdx.z * dStride;
  const float* Rb = res ? res + (size_t)blockIdx.z * dStride : nullptr;

  // A tiles: rows m0..m0+15 and m0+16..m0+31.  Per lane: row = m0+lid(+16),
  // float2 at col k + 2*hi  ->  v0 = K even slot, v1 = K odd slot.
  const float* A0 = Wt + (size_t)(m0 + lid) * K + 2 * hi;
  const float* A1 = A0 + (size_t)16 * K;
  // B tiles: cols n0+lid+16*j. Per lane: v0 = X[k+2*hi][col], v1 = X[k+2*hi+1][col]
  const float* Bp = Xb + (size_t)(2 * hi) * N + n0 + lid;

  v8f acc0[4] = {v8f{}, v8f{}, v8f{}, v8f{}};   // M-tile 0, N-tiles 0..3
  v8f acc1[4] = {v8f{}, v8f{}, v8f{}, v8f{}};   // M-tile 1, N-tiles 0..3

#pragma unroll 2
  for (int k = 0; k < K; k += 4) {
    v2f a0 = *(const v2f*)(A0 + k);
    v2f a1 = *(const v2f*)(A1 + k);
    const float* bp = Bp + (size_t)k * N;
    v2f bf[4];
#pragma unroll
    for (int j = 0; j < 4; ++j) {
      bf[j].x = bp[16 * j];
      bf[j].y = bp[N + 16 * j];
    }
#pragma unroll
    for (int j = 0; j < 4; ++j) {
      acc0[j] = wmma4(a0, bf[j], acc0[j]);
      acc1[j] = wmma4(a1, bf[j], acc1[j]);
    }
  }

  // D layout: VGPR r, lanes0-15 -> M=r, lanes16-31 -> M=r+8, N=lane%16
#pragma unroll
  for (int r = 0; r < 8; ++r) {
    int row0 = m0 + r + 8 * hi;
    int row1 = row0 + 16;
    float bv0 = bias ? bias[row0] : 0.f;
    float bv1 = bias ? bias[row1] : 0.f;
    size_t base0 = (size_t)row0 * N + n0 + lid;
    size_t base1 = (size_t)row1 * N + n0 + lid;
#pragma unroll
    for (int j = 0; j < 4; ++j) {
      size_t i0 = base0 + 16 * j;
      size_t i1 = base1 + 16 * j;
      float v0 = acc0[j][r] + bv0;
      float v1 = acc1[j][r] + bv1;
      if (Rb) { v0 += Rb[i0]; v1 += Rb[i1]; }
      Db[i0] = v0;
      Db[i1] = v1;
    }
  }
}

// ============================================================
// GroupNorm (full-sample) statistics: one block per batch.
// stats[2b] = mean, stats[2b+1] = rsqrt(var + eps). Optional x+res.
// ============================================================
__global__ __launch_bounds__(256)
void gn_stats_kernel(const float* __restrict__ x, const float* __restrict__ res,
                     float* __restrict__ stats)
{
  const int b = blockIdx.x;
  const float* xb = x + (size_t)b * PB_;
  const float* rb = res ? res + (size_t)b * PB_ : nullptr;
  float s = 0.f, s2 = 0.f;
  if (rb) {
    for (int i = threadIdx.x; i < PB_; i += 256) {
      float v = xb[i] + rb[i]; s += v; s2 += v * v;
    }
  } else {
    for (int i = threadIdx.x; i < PB_; i += 256) {
      float v = xb[i]; s += v; s2 += v * v;
    }
  }
  __shared__ float l1[256];
  __shared__ float l2[256];
  l1[threadIdx.x] = s; l2[threadIdx.x] = s2;
  __syncthreads();
  for (int off = 128; off > 0; off >>= 1) {
    if (threadIdx.x < off) {
      l1[threadIdx.x] += l1[threadIdx.x + off];
      l2[threadIdx.x] += l2[threadIdx.x + off];
    }
    __syncthreads();
  }
  if (threadIdx.x == 0) {
    float mu  = l1[0] * (1.0f / (float)PB_);
    float var = l2[0] * (1.0f / (float)PB_) - mu * mu;
    stats[2 * b]     = mu;
    stats[2 * b + 1] = rsqrtf(var + EPS_F);
  }
}

// normalized = (x(+res) - mu)*rsig * w[c] + b[c]
__global__ __launch_bounds__(256)
void gn_apply_kernel(const float* __restrict__ x, const float* __restrict__ res,
                     const float* __restrict__ stats,
                     const float* __restrict__ w, const float* __restrict__ bvec,
                     float* __restrict__ out)
{
  int idx = blockIdx.x * 256 + threadIdx.x;        // < B_*PB_ = 8388608
  int b = idx >> 19;                                // PB_ = 2^19
  int c = (idx >> 10) & (C_ - 1);                   // N_ = 2^10
  float mu = stats[2 * b], rs = stats[2 * b + 1];
  float v = x[idx];
  if (res) v += res[idx];
  out[idx] = (v - mu) * rs * w[c] + bvec[c];
}

// ============================================================
// q projection: q[b,n] = qkv_b[0] + sum_c qkv_w[0,c] * h[b,c,n]
// ============================================================
__global__ __launch_bounds__(256)
void q_proj_kernel(const float* __restrict__ h, const float* __restrict__ w0,
                   const float* __restrict__ b0, float* __restrict__ q)
{
  int idx = blockIdx.x * 256 + threadIdx.x;   // < B_*N_
  int b = idx >> 10;
  int n = idx & (N_ - 1);
  const float* hb = h + (size_t)b * PB_;
  float acc = b0[0];
#pragma unroll 4
  for (int c = 0; c < C_; ++c)
    acc = fmaf(w0[c], hb[(size_t)c * N_ + n], acc);
  q[idx] = acc;
}

// softmax over W=32 per (b,h) row; one wave per row, 8 rows per block
__global__ __launch_bounds__(256)
void softmax_w_kernel(const float* __restrict__ q, float* __restrict__ s)
{
  int row  = blockIdx.x * 8 + (threadIdx.x >> 5);   // < B_*H_
  int lane = threadIdx.x & 31;
  float v = q[row * 32 + lane];
  float m = v;
#pragma unroll
  for (int off = 16; off > 0; off >>= 1)
    m = fmaxf(m, __shfl_xor(m, off, 32));
  float e = expf(v - m);
  float ss = e;
#pragma unroll
  for (int off = 16; off > 0; off >>= 1)
    ss += __shfl_xor(ss, off, 32);
  s[row * 32 + lane] = e / ss;
}

// ctx[b,c,h] = sum_w k[b,c,h,w]*score[b,h,w];  g = relu(v)*ctx
// one wave per (b,c,h) row of 32 w's; 8 rows per block
__global__ __launch_bounds__(256)
void ctx_g_kernel(const float* __restrict__ kv, const float* __restrict__ s,
                  float* __restrict__ g)
{
  int row  = blockIdx.x * 8 + (threadIdx.x >> 5);   // < B_*C_*H_
  int lane = threadIdx.x & 31;
  int b   = row >> 14;              // C_*H_ = 16384 rows per batch
  int rem = row & 16383;
  int c   = rem >> 5;
  int hh  = rem & 31;
  size_t kvb = (size_t)b * (2 * (size_t)C_ * N_);
  size_t roff = (size_t)hh * 32 + lane;
  float kk = kv[kvb + (size_t)c * N_ + roff];
  float sc = s[((b << 5) + hh) * 32 + lane];
  float p = kk * sc;
#pragma unroll
  for (int off = 16; off > 0; off >>= 1)
    p += __shfl_xor(p, off, 32);
  float vv = kv[kvb + (size_t)(C_ + c) * N_ + roff];
  g[(size_t)b * PB_ + (size_t)c * N_ + roff] = fmaxf(vv, 0.f) * p;
}

// ---------------- workspace layout (floats) ----------------
// h:     [0,            8388608)
// big:   [8388608,      41943040)  time-shared:
//          kv   = big[0 .. 16777216)           (stages 3-6)
//          attn = big[16777216 .. 25165824)    (stages 7-9)
//          t    = big[0 .. 33554432)           (stages 10-11)
// q:     [41943040, +16384) ; scores next 16384 ; stats next 32
#define OFF_H     ((size_t)0)
#define OFF_BIG   ((size_t)8388608)
#define OFF_KV    OFF_BIG
#define OFF_ATTN  (OFF_BIG + (size_t)16777216)
#define OFF_T     OFF_BIG
#define OFF_Q     (OFF_BIG + (size_t)33554432)
#define OFF_S     (OFF_Q + (size_t)16384)
#define OFF_ST    (OFF_S + (size_t)16384)

extern "C" void kernel_launch(void* const* d_in, const int* in_sizes, int n_in,
                              void* d_out, int out_size, void* d_ws, size_t ws_size,
                              hipStream_t stream)
{
  const float* x_in  = (const float*)d_in[0];
  const float* gn1_w = (const float*)d_in[1];
  const float* gn1_b = (const float*)d_in[2];
  const float* qkv_w = (const float*)d_in[3];   // (4, 1025, 512)
  const float* qkv_b = (const float*)d_in[4];   // (4, 1025)
  const float* out_w = (const float*)d_in[5];   // (4, 512, 512)
  const float* out_b = (const float*)d_in[6];
  const float* gn2_w = (const float*)d_in[7];
  const float* gn2_b = (const float*)d_in[8];
  const float* f1_w  = (const float*)d_in[9];   // (4, 2048, 512)
  const float* f1_b  = (const float*)d_in[10];
  const float* f2_w  = (const float*)d_in[11];  // (4, 512, 2048)
  const float* f2_b  = (const float*)d_in[12];

  float* out = (float*)d_out;
  float* ws  = (float*)d_ws;

  float* h_buf    = ws + OFF_H;
  float* kv_buf   = ws + OFF_KV;
  float* attn_buf = ws + OFF_ATTN;
  float* t_buf    = ws + OFF_T;
  float* q_buf    = ws + OFF_Q;
  float* s_buf    = ws + OFF_S;
  float* stats    = ws + OFF_ST;

  for (int l = 0; l < 4; ++l) {
    const float* xcur = (l == 0) ? x_in : (const float*)out;
    const float* qw = qkv_w + (size_t)l * 1025 * C_;
    const float* qb = qkv_b + (size_t)l * 1025;

    // --- GN1 ---
    gn_stats_kernel<<<B_, 256, 0, stream>>>(xcur, nullptr, stats);
    gn_apply_kernel<<<(B_ * PB_) / 256, 256, 0, stream>>>(
        xcur, nullptr, stats, gn1_w + l * C_, gn1_b + l * C_, h_buf);

    // --- k,v projection: rows 1..1024 of qkv_w (O=1024, K=512) ---
    gemm_wmma_f32<<<dim3((2 * C_) / 64, N_ / 128, B_), 128, 0, stream>>>(
        qw + C_, h_buf, qb + 1, nullptr, kv_buf,
        2 * C_, C_, N_, (size_t)PB_, (size_t)2 * C_ * N_);

    // --- q projection + softmax over W ---
    q_proj_kernel<<<(B_ * N_) / 256, 256, 0, stream>>>(h_buf, qw, qb, q_buf);
    softmax_w_kernel<<<(B_ * H_) / 8, 256, 0, stream>>>(q_buf, s_buf);

    // --- ctx + relu(v)*ctx  (overwrites h_buf with g) ---
    ctx_g_kernel<<<(B_ * C_ * H_) / 8, 256, 0, stream>>>(kv_buf, s_buf, h_buf);

    // --- out projection: attn = out_w @ g + out_b ---
    gemm_wmma_f32<<<dim3(C_ / 64, N_ / 128, B_), 128, 0, stream>>>(
        out_w + (size_t)l * C_ * C_, h_buf, out_b + l * C_, nullptr, attn_buf,
        C_, C_, N_, (size_t)PB_, (size_t)PB_);

    // --- GN2 over (attn + y) ---
    gn_stats_kernel<<<B_, 256, 0, stream>>>(attn_buf, xcur, stats);
    gn_apply_kernel<<<(B_ * PB_) / 256, 256, 0, stream>>>(
        attn_buf, xcur, stats, gn2_w + l * C_, gn2_b + l * C_, h_buf);

    // --- FFN: t = f1_w @ h + f1_b ; x_new = f2_w @ t + f2_b + y ---
    gemm_wmma_f32<<<dim3(FFW_ / 64, N_ / 128, B_), 128, 0, stream>>>(
        f1_w + (size_t)l * FFW_ * C_, h_buf, f1_b + l * FFW_, nullptr, t_buf,
        FFW_, C_, N_, (size_t)PB_, (size_t)FFW_ * N_);

    gemm_wmma_f32<<<dim3(C_ / 64, N_ / 128, B_), 128, 0, stream>>>(
        f2_w + (size_t)l * C_ * FFW_, t_buf, f2_b + l * C_, xcur, out,
        C_, FFW_, N_, (size_t)FFW_ * N_, (size_t)PB_);
  }
}